// CausalSelfAttention_2783138808334
// MI455X (gfx1250) — compile-verified
//
#include <hip/hip_runtime.h>

// ---------------------------------------------------------------------------
// CausalSelfAttention with relative position embeddings (B=8, T=1024, C=64,
// n_head=1, fp32) for MI455X / gfx1250.
//
// Key transforms:
//   att2[b,t,s] = q[b,t,:].embk[t-s,:]  ==>  P = Q @ embk^T  (GEMM) + gather
//   y2[b,t,c]   = sum_s att[b,t,s] embv[t-s,c]
//               = (attR @ embv)[t,c] with attR[i,d] = att[i, t-d]
//     -> attR is gathered for free while loading the WMMA A operand from LDS.
//
// All matmuls use V_WMMA_F32_16X16X4_F32 (native fp32 matrix path; exact
// precision vs. the fp32 reference). Whole working set (~42 MB) is L2
// resident (192 MB), so the kernel is matrix-pipe bound, not HBM bound.
// ---------------------------------------------------------------------------

typedef float v2f __attribute__((ext_vector_type(2)));
typedef float v8f __attribute__((ext_vector_type(8)));

#define NB 8
#define NT 1024
#define NC 64

__device__ __forceinline__ v8f wmma_f32(v2f a, v2f b, v8f c) {
  // D = A(16x4) * B(4x16) + C(16x16), fp32
  return __builtin_amdgcn_wmma_f32_16x16x4_f32(false, a, false, b, (short)0, c,
                                               false, false);
}

// ---------------------------------------------------------------------------
// Kernel 1: qkv = x @ Wqkv^T + bqkv, split into q/k/v  [B,T,C] each.
// One thread per output element, float4-vectorized dot over C=64.
// ---------------------------------------------------------------------------
__global__ __launch_bounds__(256) void qkv_kernel(const float* __restrict__ x,
                                                  const float* __restrict__ Wqkv,
                                                  const float* __restrict__ bqkv,
                                                  float* __restrict__ q,
                                                  float* __restrict__ k,
                                                  float* __restrict__ v) {
  int idx = blockIdx.x * 256 + threadIdx.x;   // [0, B*T*3C)
  int bt = idx / (3 * NC);
  int j = idx % (3 * NC);
  const float4* xr = (const float4*)(x + (size_t)bt * NC);
  const float4* wr = (const float4*)(Wqkv + (size_t)j * NC);
  float s = bqkv[j];
#pragma unroll
  for (int c4 = 0; c4 < NC / 4; ++c4) {
    float4 a = xr[c4];
    float4 w = wr[c4];
    s += a.x * w.x + a.y * w.y + a.z * w.z + a.w * w.w;
  }
  int which = j >> 6;
  int c = j & (NC - 1);
  float* dst = (which == 0) ? q : ((which == 1) ? k : v);
  dst[(size_t)bt * NC + c] = s;
}

// ---------------------------------------------------------------------------
// Kernel 2: P[bt, d] = q[bt,:] . embk[d,:]   ([8192 x 64] @ [64 x 1024])
// WMMA f32 16x16x4, 4 waves per block, one 16x16 tile per wave.
// Tiles fully above the causal band (d0 > t0+15) are skipped: those entries
// are never gathered (gather uses d = t-s <= t).
// ---------------------------------------------------------------------------
__global__ __launch_bounds__(128) void p_kernel(const float* __restrict__ q,
                                                const float* __restrict__ embk,
                                                float* __restrict__ P) {
  int wave = threadIdx.x >> 5;
  int lane = threadIdx.x & 31;
  int tile = blockIdx.x * 4 + wave;           // [0, 512*64)
  int tm = tile >> 6;                         // row tile (bt/16)
  int td = tile & 63;                         // d tile
  int row0 = tm * 16;                         // global bt
  int d0 = td * 16;
  int t0 = row0 & (NT - 1);                   // t of first row (rows share b)
  if (d0 > t0 + 15) return;                   // wave-uniform causal skip

  bool hi = lane >= 16;
  int m = lane & 15;
  int off0 = hi ? 2 : 0;
  const float* arow = q + (size_t)(row0 + m) * NC;     // A[m][k]
  const float* brow = embk + (size_t)(d0 + m) * NC;    // B[k][n]=embk[d0+n][k]

  v8f acc = {};
#pragma unroll
  for (int kk = 0; kk < NC; kk += 4) {
    v2f a, b;
    a.x = arow[kk + off0];
    a.y = arow[kk + off0 + 1];
    b.x = brow[kk + off0];
    b.y = brow[kk + off0 + 1];
    acc = wmma_f32(a, b, acc);
  }
#pragma unroll
  for (int r = 0; r < 8; ++r) {
    int row = row0 + r + (hi ? 8 : 0);
    P[(size_t)row * NT + d0 + m] = acc[r];
  }
}

// ---------------------------------------------------------------------------
// Kernel 3: fused attention + output projection.
// One block (4 wave32) per (b, 16-row query tile). Full att row tile [16][1024]
// in LDS (64 KB of the 320 KB/WGP budget).
// ---------------------------------------------------------------------------
__global__ __launch_bounds__(128) void attn_kernel(
    const float* __restrict__ q, const float* __restrict__ k,
    const float* __restrict__ v, const float* __restrict__ P,
    const float* __restrict__ embv, const float* __restrict__ Wproj,
    const float* __restrict__ bproj, float* __restrict__ out) {
  __shared__ float ATT[16][NT];               // 64 KB: raw scores -> weights -> Y

  const int blk = blockIdx.x;                 // [0, 512)
  const int b = blk >> 6;
  const int t0 = (blk & 63) * 16;
  const int tid = threadIdx.x;
  const int wave = tid >> 5;
  const int lane = tid & 31;
  const bool hi = lane >= 16;
  const int m = lane & 15;
  const int off0 = hi ? 2 : 0;
  const int Send = t0 + 16;                   // causal extent (exclusive)

  const float* qb = q + ((size_t)b * NT + t0) * NC;
  const float* kb = k + (size_t)b * NT * NC;
  const float* vb = v + (size_t)b * NT * NC;
  const float* Pb = P + ((size_t)b * NT + t0) * NT;

  // ---- Phase 1: scores S = (Q K^T + P-gather) / sqrt(C), causal mask --------
  const int n_st = (t0 >> 4) + 1;             // s-tiles in causal range
  for (int st = wave; st < n_st; st += 4) {   // wave-uniform trip count
    const int s0 = st * 16;
    const float* arow = qb + (size_t)m * NC;
    const float* brow = kb + (size_t)(s0 + m) * NC;   // B[k][n]=K[s0+n][k]
    v8f acc = {};
#pragma unroll
    for (int kk = 0; kk < NC; kk += 4) {
      v2f a, bv2;
      a.x = arow[kk + off0];
      a.y = arow[kk + off0 + 1];
      bv2.x = brow[kk + off0];
      bv2.y = brow[kk + off0 + 1];
      acc = wmma_f32(a, bv2, acc);
    }
#pragma unroll
    for (int r = 0; r < 8; ++r) {
      int i = r + (hi ? 8 : 0);
      int t = t0 + i;
      int s = s0 + m;
      float sc;
      if (s <= t)
        sc = (acc[r] + Pb[(size_t)i * NT + (t - s)]) * 0.125f;
      else
        sc = -__builtin_huge_valf();
      ATT[i][s] = sc;
    }
  }
  __syncthreads();

  // ---- Phase 2: softmax (8 lanes per row, shuffle reductions) -------------
  const int row = tid >> 3;                   // row fully inside one wave
  const int slot = tid & 7;
  const int trow = t0 + row;
  float mx = -__builtin_huge_valf();
  for (int s = slot; s < Send; s += 8) mx = fmaxf(mx, ATT[row][s]);
#pragma unroll
  for (int off = 1; off < 8; off <<= 1) mx = fmaxf(mx, __shfl_xor(mx, off, 32));
  float sum = 0.0f;
  for (int s = slot; s < Send; s += 8) {
    float e = __expf(ATT[row][s] - mx);       // masked -> exp(-inf)=0
    ATT[row][s] = e;
    sum += e;
  }
#pragma unroll
  for (int off = 1; off < 8; off <<= 1) sum += __shfl_xor(sum, off, 32);
  float inv = 1.0f / sum;
  for (int s = slot; s < Send; s += 8) ATT[row][s] *= inv;
  __syncthreads();

  // ---- Phase 3: Y = att @ V + attR @ embv (wave w owns cols [16w,16w+16)) -
  const int n0 = wave * 16;
  v8f yacc = {};
  const float* bvc = vb + n0 + m;             // B[k][n] = V[s][n0+n]
#pragma unroll 4
  for (int kk = 0; kk < Send; kk += 4) {      // y1: att @ V
    v2f a, bv2;
    a.x = ATT[m][kk + off0];
    a.y = ATT[m][kk + off0 + 1];
    bv2.x = bvc[(size_t)(kk + off0) * NC];
    bv2.y = bvc[(size_t)(kk + off0 + 1) * NC];
    yacc = wmma_f32(a, bv2, yacc);
  }
  const float* bec = embv + n0 + m;           // B[k][n] = embv[d][n0+n]
  const int tr = t0 + m;                      // this lane's A row t
#pragma unroll 4
  for (int kk = 0; kk < Send; kk += 4) {      // y2: attR @ embv, attR gathered
    int d0i = kk + off0, d1i = kk + off0 + 1;
    v2f a, bv2;
    a.x = (d0i <= tr) ? ATT[m][tr - d0i] : 0.0f;
    a.y = (d1i <= tr) ? ATT[m][tr - d1i] : 0.0f;
    bv2.x = bec[(size_t)d0i * NC];
    bv2.y = bec[(size_t)d1i * NC];
    yacc = wmma_f32(a, bv2, yacc);
  }
  __syncthreads();                            // all reads of ATT done
  // Park Y tile (C-layout) into head of ATT buffer to get A-layout access.
#pragma unroll
  for (int r = 0; r < 8; ++r) {
    int i = r + (hi ? 8 : 0);
    ATT[i][n0 + m] = yacc[r];
  }
  __syncthreads();

  // ---- Phase 4: out = Y @ Wproj^T + bproj ---------------------------------
  const float* aY = &ATT[m][0];
  const float* bw = Wproj + (size_t)(n0 + m) * NC;    // B[k][n]=Wproj[n0+n][k]
  v8f pacc = {};
#pragma unroll
  for (int kk = 0; kk < NC; kk += 4) {
    v2f a, bv2;
    a.x = aY[kk + off0];
    a.y = aY[kk + off0 + 1];
    bv2.x = bw[kk + off0];
    bv2.y = bw[kk + off0 + 1];
    pacc = wmma_f32(a, bv2, pacc);
  }
  const int n = n0 + m;
  const float bias = bproj[n];
#pragma unroll
  for (int r = 0; r < 8; ++r) {
    int i = r + (hi ? 8 : 0);
    out[((size_t)b * NT + t0 + i) * NC + n] = pacc[r] + bias;
  }
}

// ---------------------------------------------------------------------------
// Launch: ws layout (floats): q[512K] | k[512K] | v[512K] | P[8M]  (~40 MB)
// ---------------------------------------------------------------------------
extern "C" void kernel_launch(void* const* d_in, const int* in_sizes, int n_in,
                              void* d_out, int out_size, void* d_ws,
                              size_t ws_size, hipStream_t stream) {
  const float* x = (const float*)d_in[0];
  const float* Wqkv = (const float*)d_in[1];
  const float* bqkv = (const float*)d_in[2];
  const float* embk = (const float*)d_in[3];
  const float* embv = (const float*)d_in[4];
  const float* Wproj = (const float*)d_in[5];
  const float* bproj = (const float*)d_in[6];
  float* out = (float*)d_out;

  float* ws = (float*)d_ws;
  float* q = ws;                                   // B*T*C = 524288
  float* k = ws + (size_t)NB * NT * NC;
  float* v = ws + (size_t)2 * NB * NT * NC;
  float* P = ws + (size_t)3 * NB * NT * NC;        // B*T*T = 8388608

  // 1) QKV projection: B*T*3C outputs, 256 threads each
  qkv_kernel<<<(NB * NT * 3 * NC) / 256, 256, 0, stream>>>(x, Wqkv, bqkv, q, k,
                                                           v);
  // 2) P = Q @ embk^T : 512*64 tiles, 4 waves/block
  p_kernel<<<(512 * 64) / 4, 128, 0, stream>>>(q, embk, P);
  // 3) fused attention + projection: one block per (b, 16-row tile)
  attn_kernel<<<NB * (NT / 16), 128, 0, stream>>>(q, k, v, P, embv, Wproj,
                                                  bproj, out);
}